// KPConv_35725537968372
// MI455X (gfx1250) — compile-verified
//
#include <hip/hip_runtime.h>

typedef __attribute__((ext_vector_type(2))) float v2f;
typedef __attribute__((ext_vector_type(8))) float v8f;

#define NQ   50000   // queries
#define MS   50000   // support points (index MS == shadow)
#define KPN  15      // kernel points
#define HN   32      // neighbors
#define CIN  64
#define COUT 128
#define KC   (KPN * CIN)   // 960 — fused reduction dim of the big GEMM
#define KPAIRS (KC / 2)    // 480 K-pairs in swizzled weights

// ---------------------------------------------------------------------------
// Kernel A: influence weights + per-kernel-point aggregation.
// Produces WF[rows][960] (= [n][k*64+c]) in workspace, fp32.
// 16 queries per 256-thread block.
// ---------------------------------------------------------------------------
__global__ __launch_bounds__(256) void kpconv_wf_kernel(
    const float* __restrict__ q_pts, const float* __restrict__ s_pts,
    const int* __restrict__ nb, const float* __restrict__ x,
    const float* __restrict__ kp, float* __restrict__ wf,
    int r0, int rows)
{
    __shared__ float kp_s[KPN * 3];
    __shared__ float a_w[16][16][HN];   // [local q][k (padded to 16)][h]
    __shared__ int   idx_s[16][HN];

    const int t = threadIdx.x;
    if (t < KPN * 3) kp_s[t] = kp[t];
    __syncthreads();

    const int n0   = r0 + blockIdx.x * 16;
    const int nEnd = r0 + rows;

    for (int p = t; p < 16 * HN; p += 256) {
        const int lq = p >> 5;
        const int h  = p & 31;
        const int n  = n0 + lq;
        int idx = MS;
        if (n < nEnd) idx = nb[(size_t)n * HN + h];
        idx_s[lq][h] = idx;
        if (idx < MS) {
            const float nx = s_pts[idx * 3 + 0] - q_pts[n * 3 + 0];
            const float ny = s_pts[idx * 3 + 1] - q_pts[n * 3 + 1];
            const float nz = s_pts[idx * 3 + 2] - q_pts[n * 3 + 2];
#pragma unroll
            for (int k = 0; k < KPN; ++k) {
                const float dx = nx - kp_s[k * 3 + 0];
                const float dy = ny - kp_s[k * 3 + 1];
                const float dz = nz - kp_s[k * 3 + 2];
                const float d2 = dx * dx + dy * dy + dz * dz;
                a_w[lq][k][h] = fmaxf(1.0f - sqrtf(d2), 0.0f);  // KP_EXTENT==1
            }
        } else {
#pragma unroll
            for (int k = 0; k < KPN; ++k) a_w[lq][k][h] = 0.0f;
        }
        a_w[lq][15][h] = 0.0f;   // pad row (k = 15)
    }
    __syncthreads();

    const int lq = t >> 4;
    const int c  = (t & 15) * 4;
    const int n  = n0 + lq;

    float4 acc[KPN] = {};

    for (int h = 0; h < HN; ++h) {
        const int idx = idx_s[lq][h];
        float4 xv = make_float4(0.0f, 0.0f, 0.0f, 0.0f);
        if (idx < MS) xv = *(const float4*)(x + (size_t)idx * CIN + c);
#pragma unroll
        for (int k = 0; k < KPN; ++k) {
            const float aw = a_w[lq][k][h];
            acc[k].x = fmaf(aw, xv.x, acc[k].x);
            acc[k].y = fmaf(aw, xv.y, acc[k].y);
            acc[k].z = fmaf(aw, xv.z, acc[k].z);
            acc[k].w = fmaf(aw, xv.w, acc[k].w);
        }
    }

    if (n < nEnd) {
        float* dst = wf + (size_t)(n - r0) * KC + c;
#pragma unroll
        for (int k = 0; k < KPN; ++k)
            *(float4*)(dst + k * CIN) = acc[k];
    }
}

// ---------------------------------------------------------------------------
// Weight pre-swizzle: Wswz[pair][c][2] = (Wflat[2p][c], Wflat[2p+1][c]).
// Makes every WMMA B-fragment a contiguous b64 and every async B stage a
// contiguous 8 KB region. Runs once per launch (480*128 pairs).
// ---------------------------------------------------------------------------
__global__ __launch_bounds__(256) void wswz_kernel(const float* __restrict__ wmat,
                                                   float* __restrict__ wswz)
{
    const int i  = blockIdx.x * 256 + threadIdx.x;  // 0 .. 480*128-1
    const int p  = i >> 7;          // K-pair index
    const int c  = i & 127;
    float2 v;
    v.x = wmat[(size_t)(2 * p + 0) * COUT + c];
    v.y = wmat[(size_t)(2 * p + 1) * COUT + c];
    *(float2*)(wswz + (size_t)i * 2) = v;
}

// ---------------------------------------------------------------------------
// CDNA5 async copy helpers (ASYNCcnt path).
// ---------------------------------------------------------------------------
__device__ __forceinline__ void async_cp16(unsigned lds_addr, const float* g) {
    asm volatile("global_load_async_to_lds_b128 %0, %1, off"
                 :: "v"(lds_addr), "v"(g) : "memory");
}
__device__ __forceinline__ void wait_async0() {
    asm volatile("s_wait_asynccnt 0x0" ::: "memory");
}
__device__ __forceinline__ v8f wmma4(v2f a, v2f b, v8f c) {
    return __builtin_amdgcn_wmma_f32_16x16x4_f32(false, a, false, b,
                                                 (short)0, c, false, false);
}

// ---------------------------------------------------------------------------
// Kernel B: out[N,128] = WF[N,960] @ W[960,128], fp32 WMMA.
// 256 threads = 8 waves = 4 row-strips x 2 col-halves; wave tile 32x64.
// K in chunks of 16, double-buffered async-to-LDS.
// Per kk-step: 2 A b64 + 4 B b64 LDS reads feed 8 WMMAs (all conflict-free).
// ---------------------------------------------------------------------------
#define KB   16              // K-chunk per stage
#define NST  (KC / KB)       // 60 stages
#define AS2  20              // Atile row stride (floats): 20r mod 64 distinct
#define BPS  288             // B pair-row stride (floats): 288 mod 64 == 32
                             //  -> lane halves hit disjoint bank sets

__global__ __launch_bounds__(256) void kpconv_gemm_kernel(
    const float* __restrict__ wf, const float* __restrict__ wswz,
    float* __restrict__ out, int r0, int rows)
{
    __shared__ float Atile[2][128 * AS2];  // [buf][row][k]
    __shared__ float Btile[2][8 * BPS];    // [buf][k-pair][c*2+j]

    const int t    = threadIdx.x;
    const int lane = t & 31;
    const int wid  = t >> 5;
    const int lrow = lane & 15;
    const int half = lane >> 4;     // K-pair select within fragment
    const int rs   = wid & 3;       // row strip (32 rows)
    const int ch   = wid >> 2;      // col half (64 cols)

    const int rowBase = blockIdx.x * 128;

    // --- staging assignment (fixed per thread) ---
    const int ar  = t >> 1;             // A row 0..127
    const int aco = (t & 1) * 8;        // A col offset 0 / 8
    int rA = rowBase + ar;
    if (rA >= rows) rA = rows - 1;      // clamp: junk rows computed, never stored
    const float* aSrc = wf + (size_t)rA * KC + aco;

    const int bp = t >> 5;              // B pair-row 0..7
    const int bo = (t & 31) * 8;        // 0..248 (floats within pair-row)
    const float* bSrc = wswz + (size_t)bp * 256 + bo;

    unsigned ldsA[2], ldsB[2];
    ldsA[0] = (unsigned)(uintptr_t)&Atile[0][ar * AS2 + aco];
    ldsA[1] = (unsigned)(uintptr_t)&Atile[1][ar * AS2 + aco];
    ldsB[0] = (unsigned)(uintptr_t)&Btile[0][bp * BPS + bo];
    ldsB[1] = (unsigned)(uintptr_t)&Btile[1][bp * BPS + bo];

    v8f acc[8] = {};   // [sub(2)][cc(4)]

    // prologue: stage 0 -> buffer 0
    async_cp16(ldsA[0],      aSrc);
    async_cp16(ldsA[0] + 16, aSrc + 4);
    async_cp16(ldsB[0],      bSrc);
    async_cp16(ldsB[0] + 16, bSrc + 4);

    for (int s = 0; s < NST; ++s) {
        const int cur = s & 1;
        wait_async0();          // my stage-s copies have landed
        __syncthreads();        // everyone's stage-s data visible

        if (s + 1 < NST) {      // kick off stage s+1 into the other buffer
            const int nxt = cur ^ 1;
            const float* aN = aSrc + (size_t)(s + 1) * KB;
            const float* bN = bSrc + (size_t)(s + 1) * (8 * 256);
            async_cp16(ldsA[nxt],      aN);
            async_cp16(ldsA[nxt] + 16, aN + 4);
            async_cp16(ldsB[nxt],      bN);
            async_cp16(ldsB[nxt] + 16, bN + 4);
        }

        const float* A0 = &Atile[cur][(rs * 32 + lrow) * AS2];
        const float* A1 = A0 + 16 * AS2;
        const float* Bp = &Btile[cur][half * BPS + 2 * (ch * 64 + lrow)];
#pragma unroll
        for (int kk = 0; kk < KB; kk += 4) {
            const int kr = kk + 2 * half;       // A K-pair {kr, kr+1}
            v2f a0 = *(const v2f*)(A0 + kr);
            v2f a1 = *(const v2f*)(A1 + kr);
#pragma unroll
            for (int cc = 0; cc < 4; ++cc) {
                // pair index kk/2 + half -> rows {kk+2*half, kk+2*half+1}
                v2f b = *(const v2f*)(Bp + (kk >> 1) * BPS + cc * 32);
                acc[cc]     = wmma4(a0, b, acc[cc]);
                acc[4 + cc] = wmma4(a1, b, acc[4 + cc]);
            }
        }
        __syncthreads();        // all reads of `cur` done before it is re-filled
    }

    // Epilogue: D layout — lane<16: VGPR j -> row j; lane>=16: VGPR j -> row 8+j.
    const int rtb = half * 8;
#pragma unroll
    for (int sub = 0; sub < 2; ++sub) {
#pragma unroll
        for (int cc = 0; cc < 4; ++cc) {
#pragma unroll
            for (int j = 0; j < 8; ++j) {
                const int L = rowBase + rs * 32 + sub * 16 + rtb + j;
                if (L < rows)
                    out[(size_t)(r0 + L) * COUT + ch * 64 + cc * 16 + lrow] =
                        acc[sub * 4 + cc][j];
            }
        }
    }
}

// ---------------------------------------------------------------------------
// Host launch. Deterministic; ws = [Wswz | WF chunk], chunked if ws is small.
// ---------------------------------------------------------------------------
extern "C" void kernel_launch(void* const* d_in, const int* in_sizes, int n_in,
                              void* d_out, int out_size, void* d_ws, size_t ws_size,
                              hipStream_t stream) {
    (void)in_sizes; (void)n_in; (void)out_size;
    const float* q_pts = (const float*)d_in[0];
    const float* s_pts = (const float*)d_in[1];
    const int*   nb    = (const int*)d_in[2];
    const float* x     = (const float*)d_in[3];
    const float* kp    = (const float*)d_in[4];
    const float* wmat  = (const float*)d_in[5];   // [K,Cin,Cout] == [960,128] flat
    float* out = (float*)d_out;

    const size_t wswzBytes = (size_t)KPAIRS * COUT * 2 * sizeof(float); // 491520
    float* wswz = (float*)d_ws;
    float* wf   = (float*)((char*)d_ws + wswzBytes);

    const size_t rowBytes = (size_t)KC * sizeof(float);
    const size_t wsLeft   = (ws_size > wswzBytes) ? ws_size - wswzBytes : 0;
    size_t maxRows = (wsLeft / rowBytes / 128) * 128;
    int chunk = (maxRows >= (size_t)NQ) ? NQ : (int)maxRows;
    if (chunk < 128) chunk = 128;   // minimal fallback

    hipLaunchKernelGGL(wswz_kernel, dim3((KPAIRS * COUT) / 256), dim3(256), 0,
                       stream, wmat, wswz);

    for (int r0 = 0; r0 < NQ; r0 += chunk) {
        const int rows = (NQ - r0 < chunk) ? (NQ - r0) : chunk;
        dim3 gA((rows + 15) / 16);
        hipLaunchKernelGGL(kpconv_wf_kernel, gA, dim3(256), 0, stream,
                           q_pts, s_pts, nb, x, kp, wf, r0, rows);
        dim3 gB((rows + 127) / 128);
        hipLaunchKernelGGL(kpconv_gemm_kernel, gB, dim3(256), 0, stream,
                           wf, wswz, out, r0, rows);
    }
}